// SignStickyPhaseReconstructor_60378650247707
// MI455X (gfx1250) — compile-verified
//
#include <hip/hip_runtime.h>
#include <stdint.h>

// Problem constants (match reference: B=256, T=131073)
#define T_LEN   131073
#define TM1     131072
#define BLOCK   1024            // 32 waves of 32 on CDNA5
#define VPT     4               // elements per thread per chunk
#define CHUNK   (BLOCK * VPT)   // 4096 elements per chunk
#define NCHUNK  (TM1 / CHUNK)   // 32 chunks per row (exact)

typedef int   vi4 __attribute__((ext_vector_type(4)));
typedef float vf4 __attribute__((ext_vector_type(4)));

__global__ __launch_bounds__(BLOCK) void sticky_sign_kernel(
    const float* __restrict__ edc,
    const int*   __restrict__ flips,
    float*       __restrict__ out)
{
    // Double-buffered edc staging (2 x 16KB) + per-wave parity slots.
    __shared__ float    sbuf[2][CHUNK];
    __shared__ unsigned wpar[2][32];

    const unsigned tid  = threadIdx.x;
    const unsigned lane = tid & 31u;
    const unsigned wave = tid >> 5;
    const int      row  = blockIdx.x;

    const float* erow = edc   + (size_t)row * T_LEN;
    const int*   frow = flips + (size_t)row * TM1;
    float*       orow = out   + (size_t)row * TM1;

    // Low 32 bits of a flat pointer into LDS == LDS byte address on gfx1250.
    const uint32_t lds0 = (uint32_t)(uintptr_t)&sbuf[0][tid * VPT];
    const uint32_t lds1 = (uint32_t)(uintptr_t)&sbuf[1][tid * VPT];
    const uint64_t gbase = (uint64_t)(uintptr_t)erow;

    // Prologue: async-copy edc chunk 0 -> sbuf[0] (GVS mode: saddr + per-lane voffset).
    {
        uint32_t voff = (uint32_t)(tid * VPT * sizeof(float));
        asm volatile("global_load_async_to_lds_b128 %0, %1, %2 offset:0"
                     :: "v"(lds0), "v"(voff), "s"(gbase) : "memory");
    }

    // Software-pipelined flips load (registers), nontemporal (read-once stream).
    vi4 fcur = __builtin_nontemporal_load((const vi4*)(frow + tid * VPT));

    unsigned carry = 0u;

    for (int c = 0; c < NCHUNK; ++c) {
        const int cb = c & 1;
        const int t0 = c * CHUNK + (int)(tid * VPT);

        // Prefetch flips two chunks ahead -> global_prefetch_b8.
        if (c + 2 < NCHUNK)
            __builtin_prefetch(frow + t0 + 2 * CHUNK, 0, 0);

        // Issue next chunk's flips load now; consumed next iteration.
        const int cn = (c + 1 < NCHUNK) ? (c + 1) : c;
        vi4 fnext = __builtin_nontemporal_load(
            (const vi4*)(frow + cn * CHUNK + tid * VPT));

        // Per-thread inclusive parities of the 4 flip bits.
        unsigned b0 = (unsigned)fcur.x & 1u;
        if (c == 0 && tid == 0) b0 = 0u;          // flips[:,0] is never applied
        const unsigned q0 = b0;
        const unsigned q1 = q0 ^ ((unsigned)fcur.y & 1u);
        const unsigned q2 = q1 ^ ((unsigned)fcur.z & 1u);
        const unsigned q3 = q2 ^ ((unsigned)fcur.w & 1u);

        // Wave-level parity scan: one ballot + popcount (wave32).
        const unsigned m    = __builtin_amdgcn_ballot_w32(q3 != 0u);
        const unsigned excl = __popc(m & ((1u << lane) - 1u)) & 1u;
        if (lane == 0) wpar[cb][wave] = __popc(m) & 1u;

        // This wave's async edc copy for chunk c must be in LDS before the barrier.
        asm volatile("s_wait_asynccnt 0x0" ::: "memory");
        __syncthreads();

        // Post-barrier: safe to overwrite the other buffer -> issue chunk c+1 copy.
        if (c + 1 < NCHUNK) {
            uint32_t voff = (uint32_t)(((c + 1) * CHUNK + tid * VPT) * sizeof(float));
            uint32_t ldsn = cb ? lds0 : lds1;
            asm volatile("global_load_async_to_lds_b128 %0, %1, %2 offset:0"
                         :: "v"(ldsn), "v"(voff), "s"(gbase) : "memory");
        }

        // Block-level scan: each lane reads one wave's parity, re-ballot gives
        // every wave the full 32-wave picture in a single mask.
        const unsigned wp    = wpar[cb][lane];
        const unsigned m2    = __builtin_amdgcn_ballot_w32(wp != 0u);
        const unsigned wexcl = __popc(m2 & ((1u << wave) - 1u)) & 1u;
        const unsigned tot   = __popc(m2) & 1u;
        const unsigned base  = carry ^ wexcl ^ excl;
        carry ^= tot;

        // edc values from LDS (16B-aligned despite odd global row stride).
        const vf4 ev = *(const vf4*)(&sbuf[cb][tid * VPT]);
        const float e4 = (tid == BLOCK - 1) ? erow[t0 + 4]
                                            : sbuf[cb][tid * VPT + 4];

        const float d0 = ev.x - ev.y;
        const float d1 = ev.y - ev.z;
        const float d2 = ev.z - ev.w;
        const float d3 = ev.w - e4;
        const float a0 = sqrtf(fmaxf(d0, 0.0f));
        const float a1 = sqrtf(fmaxf(d1, 0.0f));
        const float a2 = sqrtf(fmaxf(d2, 0.0f));
        const float a3 = sqrtf(fmaxf(d3, 0.0f));

        // Apply sign by XOR-ing parity into the sign bit (amp >= 0).
        vf4 r;
        r.x = __uint_as_float(__float_as_uint(a0) ^ ((base ^ q0) << 31));
        r.y = __uint_as_float(__float_as_uint(a1) ^ ((base ^ q1) << 31));
        r.z = __uint_as_float(__float_as_uint(a2) ^ ((base ^ q2) << 31));
        r.w = __uint_as_float(__float_as_uint(a3) ^ ((base ^ q3) << 31));

        // Streaming store, nontemporal (never re-read).
        __builtin_nontemporal_store(r, (vf4*)(orow + t0));

        fcur = fnext;
    }
}

extern "C" void kernel_launch(void* const* d_in, const int* in_sizes, int n_in,
                              void* d_out, int out_size, void* d_ws, size_t ws_size,
                              hipStream_t stream) {
    (void)n_in; (void)out_size; (void)d_ws; (void)ws_size;
    const float* edc   = (const float*)d_in[0];
    const int*   flips = (const int*)d_in[1];
    float*       out   = (float*)d_out;

    const int rows = in_sizes[0] / T_LEN;   // B = 256
    sticky_sign_kernel<<<rows, BLOCK, 0, stream>>>(edc, flips, out);
}